// MoERouter_18176301597566
// MI455X (gfx1250) — compile-verified
//
#include <hip/hip_runtime.h>

// ---------------------------------------------------------------------------
// MoE router for MI455X (gfx1250):
//   * fp32 router GEMM on the WMMA pipe (V_WMMA_F32_16X16X4_F32)
//   * W (B-matrix) staged per-block into LDS with GLOBAL_LOAD_ASYNC_TO_LDS_B128
//     (ASYNCcnt-tracked, double-buffered, retired with s_wait_asynccnt)
//   * sigmoid + grouped top-k epilogue in LDS (buffers aliased with staging)
// ---------------------------------------------------------------------------

typedef __attribute__((ext_vector_type(2))) float v2f;
typedef __attribute__((ext_vector_type(8))) float v8f;
typedef int v4i __attribute__((vector_size(16)));
typedef __attribute__((address_space(1))) v4i g_v4i;   // global int4
typedef __attribute__((address_space(3))) v4i l_v4i;   // LDS int4

#define NUM_EXPERTS   64
#define N_GROUP       8
#define EPG           8
#define TOPK_GROUP    4
#define TOPK          8
#define D_DIM         2048
#define ROWS_PER_WAVE 16
#define WAVES_PER_BLK 8
#define ROWS_PER_BLK  (ROWS_PER_WAVE * WAVES_PER_BLK)   // 128
#define K_CHUNK       64
#define NCHUNK        (D_DIM / K_CHUNK)                 // 32
#define B_STRIDE      68    // 64 + 4 pad floats: stride mod 64 banks = 4 -> conflict free
#define SEGS_PER_THD  ((NUM_EXPERTS * (K_CHUNK / 4)) / 256)   // 1024 segs / 256 thd = 4

// 16B/lane async copy global -> LDS (tracked with ASYNCcnt)
__device__ __forceinline__ void async_cp16(const float* g, float* l) {
#if __has_builtin(__builtin_amdgcn_global_load_async_to_lds_b128)
    __builtin_amdgcn_global_load_async_to_lds_b128(
        (g_v4i*)(unsigned long long)g,
        (l_v4i*)(unsigned)(unsigned long long)l,
        0, 0);
#else
    unsigned lo = (unsigned)(unsigned long long)l;   // LDS byte offset (low 32 bits)
    asm volatile("global_load_async_to_lds_b128 %0, %1, off"
                 :: "v"(lo), "v"(g) : "memory");
#endif
}

template <int N>
__device__ __forceinline__ void wait_async() {
#if __has_builtin(__builtin_amdgcn_s_wait_asynccnt)
    __builtin_amdgcn_s_wait_asynccnt(N);
#else
    asm volatile("s_wait_asynccnt %0" :: "i"(N) : "memory");
#endif
}

__global__ __launch_bounds__(256, 1) void moe_router_wmma(
    const float* __restrict__ x,      // [S, D]
    const float* __restrict__ W,      // [E, D]
    const float* __restrict__ bias,   // [E]
    int*   __restrict__ idx_out,      // [S, TOPK]
    float* __restrict__ w_out,        // [S, TOPK]
    int S)
{
    // double-buffered W staging; aliased afterwards as the 8x16x68 score spill
    __shared__ __align__(16) float smem[2][NUM_EXPERTS][B_STRIDE];   // 8704 floats
    __shared__ float lds_bias[NUM_EXPERTS];

    const int tid  = threadIdx.x;
    const int wave = tid >> 5;        // wave32
    const int lane = tid & 31;

    if (tid < NUM_EXPERTS) lds_bias[tid] = bias[tid];

    // ---- Phase 1: 16x64 fp32 GEMM tile per wave via V_WMMA_F32_16X16X4_F32 ----
    // A fragment (16x4): lanes 0..15 hold K={0,1}, lanes 16..31 K={2,3}, M = lane&15.
    // B fragment (4x16): N = lane&15 (expert), same K striping; B served from LDS.
    const int m  = lane & 15;
    const int kh = (lane >> 4) << 1;          // 0 or 2
    const int row0 = blockIdx.x * ROWS_PER_BLK + wave * ROWS_PER_WAVE;

    const float* xrow = x + (size_t)(row0 + m) * D_DIM + kh;

    v8f acc0 = {}; v8f acc1 = {}; v8f acc2 = {}; v8f acc3 = {};

    // prologue: stage chunk 0 into buffer 0
    {
        #pragma unroll
        for (int s = 0; s < SEGS_PER_THD; ++s) {
            int seg = tid + s * 256;           // 0..1023
            int e   = seg >> 4;                // expert (16 segs of 16B per expert row)
            int o   = (seg & 15) << 2;         // float offset within the 64-float chunk
            async_cp16(W + (size_t)e * D_DIM + o, &smem[0][e][o]);
        }
    }

    for (int c = 0; c < NCHUNK; ++c) {
        const int cur = c & 1;
        __syncthreads();                       // buf[cur^1] no longer read by anyone
        if (c + 1 < NCHUNK) {
            const int k0n = (c + 1) * K_CHUNK;
            #pragma unroll
            for (int s = 0; s < SEGS_PER_THD; ++s) {
                int seg = tid + s * 256;
                int e   = seg >> 4;
                int o   = (seg & 15) << 2;
                async_cp16(W + (size_t)e * D_DIM + k0n + o, &smem[cur ^ 1][e][o]);
            }
            wait_async<SEGS_PER_THD>();        // chunk c's loads (issued first) done
        } else {
            wait_async<0>();                   // final chunk: drain everything
        }
        __syncthreads();                       // all waves' chunk-c data visible

        const int    k0 = c * K_CHUNK;
        const float* bb = &smem[cur][0][0];
        __builtin_prefetch(xrow + k0 + K_CHUNK, 0, 1);   // global_prefetch_b8 for next A

        #pragma unroll 4
        for (int kk = 0; kk < K_CHUNK; kk += 4) {
            v2f a  = *(const v2f*)(xrow + k0 + kk);
            v2f b0 = *(const v2f*)(bb + (size_t)( 0 + m) * B_STRIDE + kk + kh);
            v2f b1 = *(const v2f*)(bb + (size_t)(16 + m) * B_STRIDE + kk + kh);
            v2f b2 = *(const v2f*)(bb + (size_t)(32 + m) * B_STRIDE + kk + kh);
            v2f b3 = *(const v2f*)(bb + (size_t)(48 + m) * B_STRIDE + kk + kh);
            acc0 = __builtin_amdgcn_wmma_f32_16x16x4_f32(false, a, false, b0, (short)0, acc0, false, false);
            acc1 = __builtin_amdgcn_wmma_f32_16x16x4_f32(false, a, false, b1, (short)0, acc1, false, false);
            acc2 = __builtin_amdgcn_wmma_f32_16x16x4_f32(false, a, false, b2, (short)0, acc2, false, false);
            acc3 = __builtin_amdgcn_wmma_f32_16x16x4_f32(false, a, false, b3, (short)0, acc3, false, false);
        }
    }

    // ---- Spill sigmoid(logits) to LDS (aliases the staging buffers). ----
    // C/D layout: VGPR v, lanes 0..15 -> M=v, lanes 16..31 -> M=v+8; N = lane&15.
    __syncthreads();                           // staging buffers fully consumed
    float* scores = &smem[0][0][0];
    float* wbase  = scores + (size_t)wave * ROWS_PER_WAVE * B_STRIDE;
    const int vmhi = (lane >> 4) << 3;
    #pragma unroll
    for (int v = 0; v < 8; ++v) {
        float* rp = wbase + (size_t)(v + vmhi) * B_STRIDE;
        rp[ 0 + m] = 1.0f / (1.0f + __expf(-acc0[v]));
        rp[16 + m] = 1.0f / (1.0f + __expf(-acc1[v]));
        rp[32 + m] = 1.0f / (1.0f + __expf(-acc2[v]));
        rp[48 + m] = 1.0f / (1.0f + __expf(-acc3[v]));
    }
    __syncthreads();

    // ---- Phase 2: one lane per row -> grouped top-k selection ----
    if (lane < ROWS_PER_WAVE) {
        const int row = row0 + lane;
        const float* sc = wbase + (size_t)lane * B_STRIDE;

        float gmax[N_GROUP];
        #pragma unroll
        for (int g = 0; g < N_GROUP; ++g) {
            float mx = -__builtin_inff();
            #pragma unroll
            for (int j = 0; j < EPG; ++j) {
                int e = g * EPG + j;
                mx = fmaxf(mx, sc[e] + lds_bias[e]);
            }
            gmax[g] = mx;
        }

        // top-4 groups (strict '>' ascending scan == lowest-index tie-break)
        unsigned gsel = 0u;
        #pragma unroll
        for (int t = 0; t < TOPK_GROUP; ++t) {
            float best = -__builtin_inff(); int bg = 0;
            #pragma unroll
            for (int g = 0; g < N_GROUP; ++g) {
                bool taken = (gsel >> g) & 1u;
                if (!taken && gmax[g] > best) { best = gmax[g]; bg = g; }
            }
            gsel |= 1u << bg;
        }
        unsigned long long emask = 0ull;
        #pragma unroll
        for (int g = 0; g < N_GROUP; ++g)
            if ((gsel >> g) & 1u) emask |= 0xFFull << (g * 8);

        // top-8 experts among the unmasked; capture unbiased score at pick time
        unsigned long long chosen = 0ull;
        float wsum = 0.0f;
        int   sel_i[TOPK];
        float sel_w[TOPK];
        #pragma unroll
        for (int t = 0; t < TOPK; ++t) {
            float best = -__builtin_inff(); int bi = 0; float bw = 0.0f;
            for (int e = 0; e < NUM_EXPERTS; ++e) {
                unsigned long long bit = 1ull << e;
                float raw = sc[e];
                float bv  = raw + lds_bias[e];
                bool ok = (emask & bit) && !(chosen & bit);
                if (ok && bv > best) { best = bv; bi = e; bw = raw; }
            }
            chosen |= 1ull << bi;
            sel_i[t] = bi;
            sel_w[t] = bw;
            wsum += bw;
        }

        const float inv = 1.0f / (wsum + 1e-20f);
        #pragma unroll
        for (int t = 0; t < TOPK; ++t) {
            idx_out[(size_t)row * TOPK + t] = sel_i[t];
            w_out  [(size_t)row * TOPK + t] = sel_w[t] * inv;
        }
    }
}

extern "C" void kernel_launch(void* const* d_in, const int* in_sizes, int n_in,
                              void* d_out, int out_size, void* d_ws, size_t ws_size,
                              hipStream_t stream) {
    const float* x    = (const float*)d_in[0];
    const float* W    = (const float*)d_in[1];
    const float* bias = (const float*)d_in[2];

    const int S = in_sizes[0] / D_DIM;           // 16384

    // tuple output: topk_idx (int32 [S,8]) then weights (fp32 [S,8]), flat
    int*   idx_out = (int*)d_out;
    float* w_out   = (float*)d_out + (size_t)S * TOPK;

    const int grid = S / ROWS_PER_BLK;           // 128 blocks of 256 threads (8 wave32)
    moe_router_wmma<<<grid, 256, 0, stream>>>(x, W, bias, idx_out, w_out, S);
}